// OrientationDetector_73400991088749
// MI455X (gfx1250) — compile-verified
//
#include <hip/hip_runtime.h>
#include <math.h>

#define PS      32
#define NPIX    1024       // PS*PS
#define NBINS   36
#define NTHREADS 256
#define NWAVES  8          // 256 threads / wave32
#define WH_STRIDE 38       // pad 36 -> 38 to stagger LDS banks per wave

typedef __attribute__((ext_vector_type(2))) float v2f;
typedef __attribute__((ext_vector_type(8))) float v8f;

// 3-tap replicate-pad filter matrix entry: F[a][b] over 32 indices
// F = w0*[b==max(a-1,0)] + w1*[b==a] + w2*[b==min(a+1,31)]
__device__ __forceinline__ float band3(float w0, float w1, float w2, int a, int b) {
    const int am = (a > 0)  ? a - 1 : 0;
    const int ap = (a < 31) ? a + 1 : 31;
    float v = 0.0f;
    if (b == am) v += w0;
    if (b == a)  v += w1;
    if (b == ap) v += w2;
    return v;
}

// One workgroup per patch, 8 wave32 waves.
//  Phase 1: async DMA the 4KB patch global->LDS (GLOBAL_LOAD_ASYNC_TO_LDS_B128, NT hint).
//  Phase 2: gradients as WMMA matmuls:  gy = Gy*X,  gx = X*Gx  (banded 32x32 filters).
//           8 (product,tile) tasks == 8 waves; 8 chained V_WMMA_F32_16X16X4_F32 each.
//  Phase 3: per-pixel magnitude/atan2/soft-binning; ds_add_f32 into wave-private hist.
//  Phase 4: reduce 8 wave histograms, 3-tap angular smoothing + argmax.
__global__ __launch_bounds__(NTHREADS, 4)
void orientation_kernel(const float* __restrict__ x,
                        const float* __restrict__ gx_w,
                        const float* __restrict__ gy_w,
                        const float* __restrict__ smooth_w,
                        const float* __restrict__ gk,
                        float* __restrict__ out)
{
    __shared__ float patch[NPIX];                 // staged input patch (row major)
    __shared__ float gxs[NPIX];                   // WMMA-produced d/dx
    __shared__ float gys[NPIX];                   // WMMA-produced d/dy
    __shared__ float whist[NWAVES * WH_STRIDE];   // per-wave histograms
    __shared__ float hist[NBINS];                 // reduced histogram

    const int tid  = threadIdx.x;
    const int bid  = blockIdx.x;

    // ---- Phase 1: CDNA5 async global->LDS staging (16B per lane, 4KB total) ----
    {
        // low 32 bits of the shared-aperture flat address == LDS byte offset
        unsigned ldsAddr = (unsigned)(size_t)(&patch[tid * 4]);
        const float* g = x + (size_t)bid * NPIX + (size_t)tid * 4;
        asm volatile("global_load_async_to_lds_b128 %0, %1, off th:TH_LOAD_NT"
                     :: "v"(ldsAddr), "v"(g)
                     : "memory");
    }

    // zero per-wave histograms while the async transfer is in flight
    for (int i = tid; i < NWAVES * WH_STRIDE; i += NTHREADS) whist[i] = 0.0f;

    const float wx0 = gx_w[0], wx1 = gx_w[1], wx2 = gx_w[2];
    const float wy0 = gy_w[0], wy1 = gy_w[1], wy2 = gy_w[2];

    // each wave drains its own ASYNCcnt, then workgroup barrier publishes LDS
    asm volatile("s_wait_asynccnt 0x0" ::: "memory");
    __syncthreads();

    // ---- Phase 2: gradients on the matrix pipe ----
    // task w: w<4 -> gy tile w ; w>=4 -> gx tile (w-4). Tile = 16x16 block of 32x32.
    {
        const int wv   = __builtin_amdgcn_readfirstlane(tid >> 5);   // scalar wave id
        const int lane = tid & 31;
        const int ll   = lane & 15;        // row/col within tile
        const int lh   = lane >> 4;        // half-wave select (K / M interleave)
        const int tile = wv & 3;
        const int mo   = (tile >> 1) << 4; // tile row offset: 0/16
        const int no   = (tile & 1) << 4;  // tile col offset: 0/16

        v8f acc = {};
        if (wv < 4) {
            // gy = Gy * X : A = banded filter (analytic), B = X from LDS
            #pragma unroll
            for (int ko = 0; ko < PS; ko += 4) {
                const int kx = ko + 2 * lh;
                const int m  = mo + ll;
                const int n  = no + ll;
                v2f a, b;
                a.x = band3(wy0, wy1, wy2, m, kx);
                a.y = band3(wy0, wy1, wy2, m, kx + 1);
                b.x = patch[(kx << 5) + n];
                b.y = patch[((kx + 1) << 5) + n];
                acc = __builtin_amdgcn_wmma_f32_16x16x4_f32(
                          false, a, false, b, (short)0, acc, false, false);
            }
            const int n = no + ll;
            #pragma unroll
            for (int i = 0; i < 8; ++i)
                gys[((mo + i + 8 * lh) << 5) + n] = acc[i];
        } else {
            // gx = X * Gx : A = X from LDS, B = banded filter (analytic)
            #pragma unroll
            for (int ko = 0; ko < PS; ko += 4) {
                const int kx = ko + 2 * lh;
                const int m  = mo + ll;
                const int n  = no + ll;
                v2f a, b;
                a.x = patch[(m << 5) + kx];
                a.y = patch[(m << 5) + kx + 1];
                b.x = band3(wx0, wx1, wx2, n, kx);       // Gx[k][n] == band3(n,k)
                b.y = band3(wx0, wx1, wx2, n, kx + 1);
                acc = __builtin_amdgcn_wmma_f32_16x16x4_f32(
                          false, a, false, b, (short)0, acc, false, false);
            }
            const int n = no + ll;
            #pragma unroll
            for (int i = 0; i < 8; ++i)
                gxs[((mo + i + 8 * lh) << 5) + n] = acc[i];
        }
    }
    __syncthreads();

    // ---- Phase 3: orientation -> soft histogram ----
    const float PIF    = 3.14159265358979323846f;
    const float kscale = (float)NBINS / (2.0f * PIF);
    float* wh = &whist[(tid >> 5) * WH_STRIDE];

    #pragma unroll
    for (int k = 0; k < 4; ++k) {
        const int p = tid + NTHREADS * k;          // lanes contiguous -> conflict-free LDS
        const float gxv = gxs[p];
        const float gyv = gys[p];

        const float mag = sqrtf(gxv * gxv + gyv * gyv + 1e-10f) * gk[p];
        const float ori = atan2f(gyv, gxv);            // [-pi, pi]
        const float obig = (ori + PIF) * kscale;       // [0, 36]
        const float b0f  = floorf(obig);
        const float w1   = obig - b0f;

        int b0 = (int)b0f;
        if (b0 >= NBINS) b0 -= NBINS;                  // obig == 36 edge case
        if (b0 < 0)      b0 += NBINS;
        int b1 = b0 + 1;
        if (b1 >= NBINS) b1 -= NBINS;

        atomicAdd(&wh[b0], (1.0f - w1) * mag);         // ds_add_f32
        atomicAdd(&wh[b1], w1 * mag);
    }
    __syncthreads();

    // ---- Phase 4: reduce wave histograms, smooth, argmax ----
    if (tid < NBINS) {
        float s = 0.0f;
        #pragma unroll
        for (int w = 0; w < NWAVES; ++w) s += whist[w * WH_STRIDE + tid];
        hist[tid] = s * (1.0f / (float)NPIX);          // spatial mean
    }
    __syncthreads();

    if (tid == 0) {
        const float s0 = smooth_w[0], s1 = smooth_w[1], s2 = smooth_w[2];
        float best = -INFINITY;
        int   bi   = 0;
        for (int b = 0; b < NBINS; ++b) {
            const float l  = (b > 0)         ? hist[b - 1] : 0.0f;  // zero pad
            const float rr = (b < NBINS - 1) ? hist[b + 1] : 0.0f;
            const float sm = s0 * l + s1 * hist[b] + s2 * rr;
            if (sm > best) { best = sm; bi = b; }                   // first max wins
        }
        out[bid] = 2.0f * PIF * (float)bi / (float)NBINS - PIF;
    }
}

extern "C" void kernel_launch(void* const* d_in, const int* in_sizes, int n_in,
                              void* d_out, int out_size, void* d_ws, size_t ws_size,
                              hipStream_t stream) {
    (void)n_in; (void)d_ws; (void)ws_size; (void)out_size;
    const float* x        = (const float*)d_in[0];
    const float* gx_w     = (const float*)d_in[1];
    const float* gy_w     = (const float*)d_in[2];
    const float* smooth_w = (const float*)d_in[3];
    const float* gk       = (const float*)d_in[4];
    float*       out      = (float*)d_out;

    const int B = in_sizes[0] / NPIX;   // 32768 patches
    orientation_kernel<<<B, NTHREADS, 0, stream>>>(x, gx_w, gy_w, smooth_w, gk, out);
}